// DirectGCNLayer_6468220748201
// MI455X (gfx1250) — compile-verified
//
#include <hip/hip_runtime.h>
#include <hip/hip_bf16.h>

typedef float v2f __attribute__((ext_vector_type(2)));
typedef float v8f __attribute__((ext_vector_type(8)));

#define IN_CH 128
#define OUT_CH 64

// ---------------------------------------------------------------------------
// 1) Combine weights AND transpose: Wt[c][k] = W_main[k][c] + W_shared[k][c].
//    Transposed layout makes each lane's WMMA B fragment {W[k][c], W[k+1][c]}
//    8 contiguous bytes -> single global_load_b64 with immediate offset.
// ---------------------------------------------------------------------------
__global__ void prep_weights_t(const float* __restrict__ w_main_in,
                               const float* __restrict__ w_main_out,
                               const float* __restrict__ w_shared,
                               float* __restrict__ w_in_t,    // (OUT_CH, IN_CH)
                               float* __restrict__ w_out_t) { // (OUT_CH, IN_CH)
    int i = blockIdx.x * blockDim.x + threadIdx.x;
    if (i < IN_CH * OUT_CH) {
        int k = i >> 6;   // row of original (IN_CH)
        int c = i & 63;   // col of original (OUT_CH)
        float s = w_shared[i];
        w_in_t [c * IN_CH + k] = w_main_in[i]  + s;
        w_out_t[c * IN_CH + k] = w_main_out[i] + s;
    }
}

// ---------------------------------------------------------------------------
// 2) Dual GEMM: h_in = x @ (W_main_in+W_shared), h_out = x @ (W_main_out+W_shared)
//    fp32 WMMA 16x16x4. One block = one 16-row M tile; A tile staged to LDS
//    (padded stride 132 dwords -> 16 rows spread over distinct bank groups).
//    8 waves = 8 N tiles (waves 0-3 -> h_in, waves 4-7 -> h_out).
//    Inner loop: ds_load_b64 (A) + global_load_b64 (B, immediate offsets)
//    + v_wmma_f32_16x16x4_f32; no per-iteration address VALU.
// ---------------------------------------------------------------------------
__global__ void gemm_dual_wmma(const float* __restrict__ x,
                               const float* __restrict__ w_in_t,
                               const float* __restrict__ w_out_t,
                               float* __restrict__ h_in,
                               float* __restrict__ h_out,
                               int n_nodes) {
    __shared__ __align__(16) float a_tile[16 * 132];

    const int mtile = blockIdx.x;
    const int row0  = mtile * 16;

    // Stage the 16x128 A tile into LDS as float4 (b128 load + b128 ds store).
    // 16*128 floats = 512 float4 slots; 256 threads -> 2 each.
    for (int i = threadIdx.x; i < 16 * (IN_CH / 4); i += blockDim.x) {
        int r  = i >> 5;   // / 32 float4-per-row
        int c4 = i & 31;
        float4 v = make_float4(0.f, 0.f, 0.f, 0.f);
        int grow = row0 + r;
        if (grow < n_nodes)
            v = ((const float4*)x)[(size_t)grow * (IN_CH / 4) + c4];
        *(float4*)&a_tile[r * 132 + c4 * 4] = v;   // 528B row stride: 16B aligned
    }
    __syncthreads();

    const int wave = threadIdx.x >> 5;
    const int lane = threadIdx.x & 31;
    const int r    = lane & 15;   // A row in tile / D column in tile
    const int half = lane >> 4;   // 0: K0,K1 & rows 0..7 ; 1: K2,K3 & rows 8..15

    const float* __restrict__ Wt = (wave < 4) ? w_in_t : w_out_t;
    float*       __restrict__ H  = (wave < 4) ? h_in   : h_out;
    const int nt  = wave & 3;
    const int col = nt * 16 + r;                   // output column (0..63)

    // Hoisted per-lane base pointers; loop offsets are compile-time immediates.
    const float* __restrict__ ap = &a_tile[r * 132 + 2 * half];
    const float* __restrict__ wp = Wt + (size_t)col * IN_CH + 2 * half;

    v8f acc = {};

#pragma unroll
    for (int k0 = 0; k0 < IN_CH; k0 += 4) {
        v2f a = *(const v2f*)(ap + k0);   // ds_load_b64, imm offset
        v2f b = *(const v2f*)(wp + k0);   // global_load_b64, imm offset
        acc = __builtin_amdgcn_wmma_f32_16x16x4_f32(
            /*neg_a=*/false, a, /*neg_b=*/false, b,
            /*c_mod=*/(short)0, acc, /*reuse_a=*/false, /*reuse_b=*/false);
    }

    // D layout: VGPR v -> row v + 8*half; lane&15 -> column.
#pragma unroll
    for (int v = 0; v < 8; ++v) {
        int grow = row0 + v + 8 * half;
        if (grow < n_nodes)
            H[(size_t)grow * OUT_CH + col] = acc[v];
    }
}

// ---------------------------------------------------------------------------
// 3) Initialize output with per-node-scaled bias terms:
//    out[n][c] = C_in[n]*(b_mi[c]+b_si[c]) + C_out[n]*(b_mo[c]+b_so[c])
// ---------------------------------------------------------------------------
__global__ void init_out(float* __restrict__ out,
                         const float* __restrict__ c_in,
                         const float* __restrict__ c_out,
                         const float* __restrict__ b_mi,
                         const float* __restrict__ b_mo,
                         const float* __restrict__ b_si,
                         const float* __restrict__ b_so,
                         int n_nodes) {
    int idx = blockIdx.x * blockDim.x + threadIdx.x;
    if (idx < n_nodes * OUT_CH) {
        int n = idx >> 6;
        int c = idx & 63;
        out[idx] = c_in[n] * (b_mi[c] + b_si[c]) + c_out[n] * (b_mo[c] + b_so[c]);
    }
}

// ---------------------------------------------------------------------------
// 4) Edge scatter: 64 threads per edge (one per channel). Message pre-scaled
//    by C[tgt]*w (constant per target, commutes with segment sum). Coalesced
//    256B gather of h[src], coalesced fp32 global atomics into out[tgt].
// ---------------------------------------------------------------------------
__global__ void scatter_edges(const float* __restrict__ h_in,
                              const float* __restrict__ h_out,
                              const int*   __restrict__ ei_in,
                              const float* __restrict__ ew_in,
                              const int*   __restrict__ ei_out,
                              const float* __restrict__ ew_out,
                              const float* __restrict__ c_in,
                              const float* __restrict__ c_out,
                              float* __restrict__ out,
                              int e_in, int e_out) {
    long long t = (long long)blockIdx.x * blockDim.x + threadIdx.x;
    int e = (int)(t >> 6);
    int c = (int)(t & 63);
    if (e < e_in) {
        int src = ei_in[e];
        int tgt = ei_in[e_in + e];
        float m = c_in[tgt] * ew_in[e] * h_in[(size_t)src * OUT_CH + c];
        unsafeAtomicAdd(&out[(size_t)tgt * OUT_CH + c], m);   // global_atomic_add_f32
    } else {
        e -= e_in;
        if (e < e_out) {
            int src = ei_out[e];
            int tgt = ei_out[e_out + e];
            float m = c_out[tgt] * ew_out[e] * h_out[(size_t)src * OUT_CH + c];
            unsafeAtomicAdd(&out[(size_t)tgt * OUT_CH + c], m);
        }
    }
}

// ---------------------------------------------------------------------------
extern "C" void kernel_launch(void* const* d_in, const int* in_sizes, int n_in,
                              void* d_out, int out_size, void* d_ws, size_t ws_size,
                              hipStream_t stream) {
    const float* x          = (const float*)d_in[0];
    const float* w_main_in  = (const float*)d_in[1];
    const float* w_main_out = (const float*)d_in[2];
    const float* w_shared   = (const float*)d_in[3];
    const float* b_mi       = (const float*)d_in[4];
    const float* b_mo       = (const float*)d_in[5];
    const float* b_si       = (const float*)d_in[6];
    const float* b_so       = (const float*)d_in[7];
    const float* c_in       = (const float*)d_in[8];
    const float* c_out      = (const float*)d_in[9];
    const int*   ei_in      = (const int*)  d_in[10];
    const float* ew_in      = (const float*)d_in[11];
    const int*   ei_out     = (const int*)  d_in[12];
    const float* ew_out     = (const float*)d_in[13];
    float* out = (float*)d_out;

    const int n_nodes = in_sizes[8];   // C_in_vec is (N,1)
    const int e_in    = in_sizes[11];  // edge_weight_in
    const int e_out   = in_sizes[13];  // edge_weight_out

    // Workspace layout (floats): W_in_t | W_out_t | h_in | h_out
    float* ws      = (float*)d_ws;
    float* w_in_t  = ws;
    float* w_out_t = w_in_t  + IN_CH * OUT_CH;
    float* h_in    = w_out_t + IN_CH * OUT_CH;
    float* h_out   = h_in    + (size_t)n_nodes * OUT_CH;

    // 1) combine + transpose weights
    prep_weights_t<<<(IN_CH * OUT_CH + 255) / 256, 256, 0, stream>>>(
        w_main_in, w_main_out, w_shared, w_in_t, w_out_t);

    // 2) dual GEMM via WMMA f32 16x16x4
    const int mtiles = (n_nodes + 15) / 16;
    gemm_dual_wmma<<<mtiles, 256, 0, stream>>>(x, w_in_t, w_out_t, h_in, h_out, n_nodes);

    // 3) bias/init pass
    const int total = n_nodes * OUT_CH;
    init_out<<<(total + 255) / 256, 256, 0, stream>>>(
        out, c_in, c_out, b_mi, b_mo, b_si, b_so, n_nodes);

    // 4) fused propagate + combine via fp32 global atomics
    const long long scatter_threads = ((long long)e_in + e_out) * OUT_CH;
    const int scatter_blocks = (int)((scatter_threads + 255) / 256);
    scatter_edges<<<scatter_blocks, 256, 0, stream>>>(
        h_in, h_out, ei_in, ew_in, ei_out, ew_out, c_in, c_out, out, e_in, e_out);
}